// TransformerBlock_31808527794229
// MI455X (gfx1250) — compile-verified
//
#include <hip/hip_runtime.h>
#include <math.h>

typedef __attribute__((ext_vector_type(2))) float v2f;
typedef __attribute__((ext_vector_type(8))) float v8f;

#define E_   256
#define H_   8
#define D_   32
#define FF_  768
#define NE_  6
#define B_   4
#define S_   2048
#define N_   (B_*S_)            // 8192 tokens
#define CAP_ (2*N_ + NE_*16)    // 16480 padded expert slots
#define TILES_ (CAP_/16)        // 1030

// ---------------- WMMA helper: D = A(16x4) * B(4x16) + C, fp32 ----------------
__device__ __forceinline__ v8f wmma_f32(v2f a, v2f b, v8f c) {
  return __builtin_amdgcn_wmma_f32_16x16x4_f32(false, a, false, b, (short)0, c, false, false);
}

// ---------------- init scratch ----------------
__global__ void k_init(int* __restrict__ slots, int* __restrict__ counts, int* __restrict__ cursors) {
  int i = blockIdx.x * 256 + threadIdx.x;
  if (i < CAP_) slots[i] = -1;
  if (i < NE_) { counts[i] = 0; cursors[i] = 0; }
}

// ---------------- RMSNorm: one wave per row of 256 ----------------
__global__ void k_rmsnorm(const float* __restrict__ x, const float* __restrict__ w,
                          float* __restrict__ out) {
  int wave = threadIdx.x >> 5;
  int lane = threadIdx.x & 31;
  int row  = blockIdx.x * 8 + wave;
  const float* xr = x + (size_t)row * E_;
  float4 a = *(const float4*)(xr + lane * 8);
  float4 b = *(const float4*)(xr + lane * 8 + 4);
  float ss = a.x*a.x + a.y*a.y + a.z*a.z + a.w*a.w
           + b.x*b.x + b.y*b.y + b.z*b.z + b.w*b.w;
  for (int off = 16; off; off >>= 1) ss += __shfl_xor(ss, off, 32);
  float r = rsqrtf(ss * (1.0f / E_) + 1e-6f);
  float4 w1 = *(const float4*)(w + lane * 8);
  float4 w2 = *(const float4*)(w + lane * 8 + 4);
  float* orow = out + (size_t)row * E_;
  float4 o1, o2;
  o1.x = a.x*r*w1.x; o1.y = a.y*r*w1.y; o1.z = a.z*r*w1.z; o1.w = a.w*r*w1.w;
  o2.x = b.x*r*w2.x; o2.y = b.y*r*w2.y; o2.z = b.z*r*w2.z; o2.w = b.w*r*w2.w;
  *(float4*)(orow + lane * 8)     = o1;
  *(float4*)(orow + lane * 8 + 4) = o2;
}

// ---------------- Generic GEMM: C[M,N] = A[M,K] @ B[K,N]; 64x64 tile / wave ----------------
__global__ __launch_bounds__(32) void k_gemm(const float* __restrict__ A,
                                             const float* __restrict__ Bm,
                                             float* __restrict__ C, int K, int Nn) {
  int l = threadIdx.x, lane16 = l & 15, half = l >> 4;
  int m0 = blockIdx.x * 64, n0 = blockIdx.y * 64;
  v8f acc[4][4] = {};
  for (int k0 = 0; k0 < K; k0 += 4) {
    v2f a[4];
#pragma unroll
    for (int mt = 0; mt < 4; ++mt)
      a[mt] = *(const v2f*)(A + (size_t)(m0 + mt*16 + lane16) * K + k0 + 2*half);
    v2f b[4];
    int kk = k0 + 2 * half;
#pragma unroll
    for (int nt = 0; nt < 4; ++nt) {
      const float* bp = Bm + (size_t)kk * Nn + n0 + nt*16 + lane16;
      v2f t; t.x = bp[0]; t.y = bp[Nn]; b[nt] = t;
    }
#pragma unroll
    for (int mt = 0; mt < 4; ++mt)
#pragma unroll
      for (int nt = 0; nt < 4; ++nt)
        acc[mt][nt] = wmma_f32(a[mt], b[nt], acc[mt][nt]);
  }
#pragma unroll
  for (int mt = 0; mt < 4; ++mt)
#pragma unroll
    for (int nt = 0; nt < 4; ++nt)
#pragma unroll
      for (int r = 0; r < 8; ++r)
        C[(size_t)(m0 + mt*16 + r + 8*half) * Nn + n0 + nt*16 + lane16] = acc[mt][nt][r];
}

// ---------------- GEMM with fused residual add: C = res + A@B ----------------
__global__ __launch_bounds__(32) void k_gemm_resid(const float* __restrict__ A,
                                                   const float* __restrict__ Bm,
                                                   const float* __restrict__ res,
                                                   float* __restrict__ C, int K, int Nn) {
  int l = threadIdx.x, lane16 = l & 15, half = l >> 4;
  int m0 = blockIdx.x * 64, n0 = blockIdx.y * 64;
  v8f acc[4][4] = {};
  for (int k0 = 0; k0 < K; k0 += 4) {
    v2f a[4];
#pragma unroll
    for (int mt = 0; mt < 4; ++mt)
      a[mt] = *(const v2f*)(A + (size_t)(m0 + mt*16 + lane16) * K + k0 + 2*half);
    v2f b[4];
    int kk = k0 + 2 * half;
#pragma unroll
    for (int nt = 0; nt < 4; ++nt) {
      const float* bp = Bm + (size_t)kk * Nn + n0 + nt*16 + lane16;
      v2f t; t.x = bp[0]; t.y = bp[Nn]; b[nt] = t;
    }
#pragma unroll
    for (int mt = 0; mt < 4; ++mt)
#pragma unroll
      for (int nt = 0; nt < 4; ++nt)
        acc[mt][nt] = wmma_f32(a[mt], b[nt], acc[mt][nt]);
  }
#pragma unroll
  for (int mt = 0; mt < 4; ++mt)
#pragma unroll
    for (int nt = 0; nt < 4; ++nt)
#pragma unroll
      for (int r = 0; r < 8; ++r) {
        size_t idx = (size_t)(m0 + mt*16 + r + 8*half) * Nn + n0 + nt*16 + lane16;
        C[idx] = res[idx] + acc[mt][nt][r];
      }
}

// ---------------- RoPE in-place on q,k inside qkv [N, 3*E] ----------------
__global__ void k_rope(float* __restrict__ qkv) {
  int idx = blockIdx.x * 256 + threadIdx.x;      // N*H*16 pairs
  if (idx >= N_ * H_ * 16) return;
  int i = idx & 15;
  int h = (idx >> 4) & 7;
  int n = idx >> 7;
  int pos = n & (S_ - 1);
  int d0 = 2 * i, d1 = 2 * i + 1;
  const float lf = 9.210340371976184f; // log(10000)
  float f0 = (float)pos * expf(-((float)(d0 & 15)) * (1.0f/16.0f) * lf);
  float f1 = (float)pos * expf(-((float)(d1 & 15)) * (1.0f/16.0f) * lf);
  float c0, s0, c1, s1;
  sincosf(f0, &s0, &c0);
  sincosf(f1, &s1, &c1);
  float* base = qkv + (size_t)n * (3*E_) + h * D_;
  float q0 = base[d0], q1 = base[d1];
  base[d0] = q0 * c0 - q1 * s0;     // even: x*cos + (-x_odd)*sin
  base[d1] = q1 * c1 + q0 * s1;     // odd:  x*cos + ( x_even)*sin
  float* kb = base + E_;
  float k0 = kb[d0], k1 = kb[d1];
  kb[d0] = k0 * c0 - k1 * s0;
  kb[d1] = k1 * c1 + k0 * s1;
}

// ---------------- Flash attention: 1 wave per (b,h, 16-query tile) ----------------
__global__ __launch_bounds__(32) void k_flash(const float* __restrict__ qkv,
                                              float* __restrict__ attn) {
  __shared__ float pt[16][18];
  int l = threadIdx.x, lane16 = l & 15, half = l >> 4;
  int qt = blockIdx.x;
  int bh = blockIdx.y;
  int b = bh >> 3, h = bh & 7;
  const float* base = qkv + (size_t)b * S_ * (3*E_);
  int qcol = h * D_, kcol = E_ + h * D_, vcol = 2*E_ + h * D_;
  const float scale = 0.1767766952966369f; // 32^-0.5

  v2f qa[8];
#pragma unroll
  for (int kk = 0; kk < 8; ++kk) {
    const float* p = base + (size_t)(qt*16 + lane16) * (3*E_) + qcol + 4*kk + 2*half;
    v2f t = *(const v2f*)p;
    t.x *= scale; t.y *= scale;
    qa[kk] = t;
  }
  v8f o0 = {}, o1 = {};
  float rmax[8], rsum[8];
#pragma unroll
  for (int r = 0; r < 8; ++r) { rmax[r] = -3.0e38f; rsum[r] = 0.f; }

  for (int j = 0; j <= qt; ++j) {
    v8f s = {};
#pragma unroll
    for (int kk = 0; kk < 8; ++kk) {
      const float* p = base + (size_t)(j*16 + lane16) * (3*E_) + kcol + 4*kk + 2*half;
      v2f kb = *(const v2f*)p;
      s = wmma_f32(qa[kk], kb, s);
    }
    if (j == qt) {
#pragma unroll
      for (int r = 0; r < 8; ++r) {
        int row = r + 8 * half;
        if (lane16 > row) s[r] = -3.0e38f;   // causal: col > row
      }
    }
#pragma unroll
    for (int r = 0; r < 8; ++r) {
      float v = s[r];
      v = fmaxf(v, __shfl_xor(v, 1, 32));
      v = fmaxf(v, __shfl_xor(v, 2, 32));
      v = fmaxf(v, __shfl_xor(v, 4, 32));
      v = fmaxf(v, __shfl_xor(v, 8, 32));
      float mnew  = fmaxf(rmax[r], v);
      float alpha = __expf(rmax[r] - mnew);
      rmax[r] = mnew;
      float p = __expf(s[r] - mnew);
      s[r] = p;
      float ps = p;
      ps += __shfl_xor(ps, 1, 32);
      ps += __shfl_xor(ps, 2, 32);
      ps += __shfl_xor(ps, 4, 32);
      ps += __shfl_xor(ps, 8, 32);
      rsum[r] = rsum[r] * alpha + ps;
      o0[r] *= alpha; o1[r] *= alpha;
    }
    // transpose P (C-layout -> A-layout) through LDS
#pragma unroll
    for (int r = 0; r < 8; ++r) pt[r + 8*half][lane16] = s[r];
    __syncthreads();
#pragma unroll
    for (int kk = 0; kk < 4; ++kk) {
      v2f pa;
      pa.x = pt[lane16][4*kk + 2*half];
      pa.y = pt[lane16][4*kk + 2*half + 1];
      int krow = j*16 + 4*kk + 2*half;
      const float* vp0 = base + (size_t)krow * (3*E_) + vcol + lane16;
      v2f vb0; vb0.x = vp0[0]; vb0.y = vp0[3*E_];
      o0 = wmma_f32(pa, vb0, o0);
      const float* vp1 = vp0 + 16;
      v2f vb1; vb1.x = vp1[0]; vb1.y = vp1[3*E_];
      o1 = wmma_f32(pa, vb1, o1);
    }
    __syncthreads();
  }
#pragma unroll
  for (int r = 0; r < 8; ++r) {
    float inv = 1.f / rsum[r];
    int token = b * S_ + qt*16 + r + 8*half;
    float* op = attn + (size_t)token * E_ + h * D_ + lane16;
    op[0]  = o0[r] * inv;
    op[16] = o1[r] * inv;
  }
}

// ---------------- Gate: softmax over 6, top-2, re-softmax weights ----------------
__global__ void k_gate(const float* __restrict__ xn2, const float* __restrict__ gw,
                       int* __restrict__ gidx, float* __restrict__ gwt,
                       int* __restrict__ counts, float* __restrict__ imp) {
  __shared__ float sprob[8][NE_];
  int warp = threadIdx.x >> 5, lane = threadIdx.x & 31;
  int t = blockIdx.x * 8 + warp;
  const float* xr = xn2 + (size_t)t * E_;
  float acc[NE_] = {};
  for (int j = 0; j < 8; ++j) {
    float xv = xr[lane * 8 + j];
    const float* grow = gw + (size_t)(lane * 8 + j) * NE_;
#pragma unroll
    for (int e = 0; e < NE_; ++e) acc[e] += xv * grow[e];
  }
#pragma unroll
  for (int e = 0; e < NE_; ++e)
    for (int off = 16; off; off >>= 1) acc[e] += __shfl_xor(acc[e], off, 32);
  if (lane == 0) {
    float mx = acc[0];
    for (int e = 1; e < NE_; ++e) mx = fmaxf(mx, acc[e]);
    float p[NE_]; float Z = 0.f;
    for (int e = 0; e < NE_; ++e) { p[e] = __expf(acc[e] - mx); Z += p[e]; }
    float invZ = 1.f / Z;
    for (int e = 0; e < NE_; ++e) { p[e] *= invZ; sprob[warp][e] = p[e]; }
    int bi = 0; float bp = p[0];
    for (int e = 1; e < NE_; ++e) if (p[e] > bp) { bp = p[e]; bi = e; }
    int si = 0; float sp = -1.f;
    for (int e = 0; e < NE_; ++e) if (e != bi && p[e] > sp) { sp = p[e]; si = e; }
    float m2 = fmaxf(bp, sp);
    float w0 = __expf(bp - m2), w1 = __expf(sp - m2);
    float iw = 1.f / (w0 + w1);
    gidx[2*t] = bi; gidx[2*t + 1] = si;
    gwt[2*t] = w0 * iw; gwt[2*t + 1] = w1 * iw;
    atomicAdd(&counts[bi], 1);
    atomicAdd(&counts[si], 1);
  }
  __syncthreads();
  if (threadIdx.x < NE_) {
    float s = 0.f;
    for (int wv = 0; wv < 8; ++wv) s += sprob[wv][threadIdx.x];
    imp[blockIdx.x * NE_ + threadIdx.x] = s;
  }
}

// ---------------- Offsets (padded to 16) + aux loss (deterministic reduce) ----------------
__global__ void k_offs_aux(const int* __restrict__ counts, int* __restrict__ offs,
                           const float* __restrict__ imp, float* __restrict__ aux_out) {
  __shared__ float isum[NE_];
  int tid = threadIdx.x;
  if (tid < NE_) {
    float s = 0.f;
    for (int bk = 0; bk < N_ / 8; ++bk) s += imp[bk * NE_ + tid];
    isum[tid] = s;
  }
  __syncthreads();
  if (tid == 0) {
    int off = 0;
    for (int e = 0; e < NE_; ++e) { offs[e] = off; off += ((counts[e] + 15) >> 4) << 4; }
    offs[NE_] = off;
    float aux = 0.f;
    for (int e = 0; e < NE_; ++e)
      aux += (isum[e] * (1.f / N_)) * ((float)counts[e] * (1.f / N_));
    aux_out[0] = (float)NE_ * aux;
  }
}

// ---------------- Scatter tokens into padded per-expert slot lists ----------------
__global__ void k_scatter(const int* __restrict__ gidx, const int* __restrict__ offs,
                          int* __restrict__ cursors, int* __restrict__ slots,
                          int* __restrict__ slotof) {
  int t = blockIdx.x * 256 + threadIdx.x;
  if (t >= N_) return;
#pragma unroll
  for (int k = 0; k < 2; ++k) {
    int e = gidx[2*t + k];
    int pos = atomicAdd(&cursors[e], 1);
    int slot = offs[e] + pos;
    slots[slot] = t;
    slotof[2*t + k] = slot;
  }
}

// ---------------- MoE up: h = silu(Xg @ Wg[e]) * (Xg @ Wu[e]); 16x64 tile / wave ----------------
__global__ __launch_bounds__(32) void k_moe_gu(const float* __restrict__ xn2,
                                               const float* __restrict__ Wg,
                                               const float* __restrict__ Wu,
                                               const int* __restrict__ slots,
                                               const int* __restrict__ offs,
                                               float* __restrict__ hbuf) {
  int l = threadIdx.x, lane16 = l & 15, half = l >> 4;
  int base = blockIdx.x * 16;
  int total = offs[NE_];
  if (base >= total) return;
  int e = 0;
#pragma unroll
  for (int ee = 1; ee < NE_; ++ee) if (base >= offs[ee]) e = ee;
  int tok = slots[base + lane16];
  if (tok < 0) tok = 0;                 // padding slot: compute garbage, never gathered
  const float* Arow = xn2 + (size_t)tok * E_;
  int n0 = blockIdx.y * 64;
  const float* Wge = Wg + (size_t)e * E_ * FF_;
  const float* Wue = Wu + (size_t)e * E_ * FF_;
  v8f g[4] = {}, u[4] = {};
  for (int k0 = 0; k0 < E_; k0 += 4) {
    v2f a = *(const v2f*)(Arow + k0 + 2*half);
    int kk = k0 + 2 * half;
#pragma unroll
    for (int nt = 0; nt < 4; ++nt) {
      int col = n0 + nt*16 + lane16;
      const float* pg = Wge + (size_t)kk * FF_ + col;
      v2f bg; bg.x = pg[0]; bg.y = pg[FF_];
      g[nt] = wmma_f32(a, bg, g[nt]);
      const float* pu = Wue + (size_t)kk * FF_ + col;
      v2f bu; bu.x = pu[0]; bu.y = pu[FF_];
      u[nt] = wmma_f32(a, bu, u[nt]);
    }
  }
#pragma unroll
  for (int nt = 0; nt < 4; ++nt)
#pragma unroll
    for (int r = 0; r < 8; ++r) {
      float gv = g[nt][r];
      float hv = gv / (1.f + __expf(-gv)) * u[nt][r];
      hbuf[(size_t)(base + r + 8*half) * FF_ + n0 + nt*16 + lane16] = hv;
    }
}

// ---------------- MoE down: y = h @ Wd[e]; 16x64 tile / wave ----------------
__global__ __launch_bounds__(32) void k_moe_down(const float* __restrict__ hbuf,
                                                 const float* __restrict__ Wd,
                                                 const int* __restrict__ offs,
                                                 float* __restrict__ ybuf) {
  int l = threadIdx.x, lane16 = l & 15, half = l >> 4;
  int base = blockIdx.x * 16;
  int total = offs[NE_];
  if (base >= total) return;
  int e = 0;
#pragma unroll
  for (int ee = 1; ee < NE_; ++ee) if (base >= offs[ee]) e = ee;
  const float* Arow = hbuf + (size_t)(base + lane16) * FF_;
  const float* Wde = Wd + (size_t)e * FF_ * E_;
  int n0 = blockIdx.y * 64;
  v8f acc[4] = {};
  for (int k0 = 0; k0 < FF_; k0 += 4) {
    v2f a = *(const v2f*)(Arow + k0 + 2*half);
    int kk = k0 + 2 * half;
#pragma unroll
    for (int nt = 0; nt < 4; ++nt) {
      const float* bp = Wde + (size_t)kk * E_ + n0 + nt*16 + lane16;
      v2f bb; bb.x = bp[0]; bb.y = bp[E_];
      acc[nt] = wmma_f32(a, bb, acc[nt]);
    }
  }
#pragma unroll
  for (int nt = 0; nt < 4; ++nt)
#pragma unroll
    for (int r = 0; r < 8; ++r)
      ybuf[(size_t)(base + r + 8*half) * E_ + n0 + nt*16 + lane16] = acc[nt][r];
}

// ---------------- Final: out = xres + w0*y[slot0] + w1*y[slot1] ----------------
__global__ void k_final(const float* __restrict__ xres, const float* __restrict__ ybuf,
                        const int* __restrict__ slotof, const float* __restrict__ gwt,
                        float* __restrict__ out) {
  int tid = blockIdx.x * 256 + threadIdx.x;    // N*64 threads, float4 each
  int t = tid >> 6;
  int c = (tid & 63) * 4;
  int s0 = slotof[2*t], s1 = slotof[2*t + 1];
  float w0 = gwt[2*t], w1 = gwt[2*t + 1];
  float4 xr = *(const float4*)(xres + (size_t)t * E_ + c);
  float4 y0 = *(const float4*)(ybuf + (size_t)s0 * E_ + c);
  float4 y1 = *(const float4*)(ybuf + (size_t)s1 * E_ + c);
  float4 o;
  o.x = xr.x + w0 * y0.x + w1 * y1.x;
  o.y = xr.y + w0 * y0.y + w1 * y1.y;
  o.z = xr.z + w0 * y0.z + w1 * y1.z;
  o.w = xr.w + w0 * y0.w + w1 * y1.w;
  *(float4*)(out + (size_t)t * E_ + c) = o;
}

extern "C" void kernel_launch(void* const* d_in, const int* in_sizes, int n_in,
                              void* d_out, int out_size, void* d_ws, size_t ws_size,
                              hipStream_t stream) {
  (void)in_sizes; (void)n_in; (void)out_size; (void)ws_size;
  const float* x     = (const float*)d_in[0];
  const float* Wqkv  = (const float*)d_in[1];
  const float* Wout  = (const float*)d_in[2];
  const float* rms1  = (const float*)d_in[3];
  const float* rms2  = (const float*)d_in[4];
  const float* gateW = (const float*)d_in[5];
  const float* Wg    = (const float*)d_in[6];
  const float* Wu    = (const float*)d_in[7];
  const float* Wd    = (const float*)d_in[8];
  float* out = (float*)d_out;

  // workspace layout
  float* f     = (float*)d_ws;
  float* xn1   = f;                             // N*E
  float* qkv   = xn1  + (size_t)N_ * E_;        // N*3E
  float* attn  = qkv  + (size_t)N_ * 3 * E_;    // N*E
  float* xres  = attn + (size_t)N_ * E_;        // N*E
  float* xn2   = xres + (size_t)N_ * E_;        // N*E
  float* hbuf  = xn2  + (size_t)N_ * E_;        // CAP*FF
  float* ybuf  = hbuf + (size_t)CAP_ * FF_;     // CAP*E
  float* gatew = ybuf + (size_t)CAP_ * E_;      // N*2
  float* imp   = gatew + (size_t)N_ * 2;        // (N/8)*NE
  int* ibase   = (int*)(imp + (size_t)(N_/8) * NE_);
  int* gidx    = ibase;                         // N*2
  int* slots   = gidx  + N_ * 2;                // CAP
  int* slotof  = slots + CAP_;                  // N*2
  int* counts  = slotof + N_ * 2;               // NE
  int* cursors = counts + NE_;                  // NE
  int* offs    = cursors + NE_;                 // NE+1

  dim3 b32(32);

  k_init      <<<(CAP_ + 255) / 256, 256, 0, stream>>>(slots, counts, cursors);
  k_rmsnorm   <<<N_ / 8, 256, 0, stream>>>(x, rms1, xn1);
  k_gemm      <<<dim3(N_ / 64, (3 * E_) / 64), b32, 0, stream>>>(xn1, Wqkv, qkv, E_, 3 * E_);
  k_rope      <<<(N_ * H_ * 16) / 256, 256, 0, stream>>>(qkv);
  k_flash     <<<dim3(S_ / 16, B_ * H_), b32, 0, stream>>>(qkv, attn);
  k_gemm_resid<<<dim3(N_ / 64, E_ / 64), b32, 0, stream>>>(attn, Wout, x, xres, E_, E_);
  k_rmsnorm   <<<N_ / 8, 256, 0, stream>>>(xres, rms2, xn2);
  k_gate      <<<N_ / 8, 256, 0, stream>>>(xn2, gateW, gidx, gatew, counts, imp);
  k_offs_aux  <<<1, 64, 0, stream>>>(counts, offs, imp, out + (size_t)N_ * E_);
  k_scatter   <<<N_ / 256, 256, 0, stream>>>(gidx, offs, cursors, slots, slotof);
  k_moe_gu    <<<dim3(TILES_, FF_ / 64), b32, 0, stream>>>(xn2, Wg, Wu, slots, offs, hbuf);
  k_moe_down  <<<dim3(TILES_, E_ / 64), b32, 0, stream>>>(hbuf, Wd, offs, ybuf);
  k_final     <<<(N_ * 64) / 256, 256, 0, stream>>>(xres, ybuf, slotof, gatew, out);
}